// ModelCLPM_13829794693585
// MI455X (gfx1250) — compile-verified
//
#include <hip/hip_runtime.h>
#include <math.h>

#define PENALTY_F   10.0f
#define N_CP        17
#define N_SEG       16          // N_CP - 1
#define N_SIMPSON   64
#define MAX_FS      4000        // LDS sizing for receiver positions
#define LOG2E_F     1.44269504088896340736f

typedef float v2f __attribute__((ext_vector_type(2)));
typedef float v8f __attribute__((ext_vector_type(8)));

#if defined(__has_builtin)
#  if __has_builtin(__builtin_amdgcn_wmma_f32_16x16x4_f32)
#    define HAVE_WMMA_F32X4 1
#  endif
#  if __has_builtin(__builtin_amdgcn_exp2f)
#    define HAVE_EXP2 1
#  endif
#endif
#ifndef HAVE_WMMA_F32X4
#  define HAVE_WMMA_F32X4 0
#endif
#ifndef HAVE_EXP2
#  define HAVE_EXP2 0
#endif

// exp of an UNSCALED argument (helper for non-matrix paths)
__device__ __forceinline__ float fast_exp(float x) {
#if HAVE_EXP2
    return __builtin_amdgcn_exp2f(x * LOG2E_F);
#else
    return __expf(x);
#endif
}

// exp2 of a PRE-SCALED (by log2e) argument (matrix path)
__device__ __forceinline__ float fast_exp2(float x) {
#if HAVE_EXP2
    return __builtin_amdgcn_exp2f(x);
#else
    return __expf(x);   // matrix path leaves A unscaled in this configuration
#endif
}

// ---------------------------------------------------------------------------
// prior = sum over (bs, 16) of sqrt(dx^2+dy^2); raw sum written to slot[0]
// ---------------------------------------------------------------------------
__global__ void prior_kernel(const float* __restrict__ Z,
                             const int* __restrict__ nodes,
                             int bs, float* __restrict__ slot) {
    float v = 0.0f;
    int total = bs * N_SEG;
    for (int idx = threadIdx.x; idx < total; idx += blockDim.x) {
        int i = idx / N_SEG;
        int seg = idx - i * N_SEG;
        const float* zp = Z + (size_t)nodes[i] * (2 * N_CP);
        float dx = zp[seg + 1] - zp[seg];
        float dy = zp[N_CP + seg + 1] - zp[N_CP + seg];
        v += sqrtf(dx * dx + dy * dy);
    }
    __shared__ float red[256];
    red[threadIdx.x] = v;
    __syncthreads();
    for (int s = 128; s > 0; s >>= 1) {
        if (threadIdx.x < s) red[threadIdx.x] += red[threadIdx.x + s];
        __syncthreads();
    }
    if (threadIdx.x == 0) slot[0] = red[0];
}

// ---------------------------------------------------------------------------
// fllt: per-event log-intensity, block tree-reduced into partials[block]
// ---------------------------------------------------------------------------
__global__ void fllt_kernel(const float* __restrict__ Z,
                            const float* __restrict__ beta,
                            const float* __restrict__ ts,
                            const int* __restrict__ snd,
                            const int* __restrict__ rcv,
                            int n_events, float seg_inv,
                            float* __restrict__ partials) {
    int i = blockIdx.x * blockDim.x + threadIdx.x;
    float v = 0.0f;
    if (i < n_events) {
        float u = ts[i] * seg_inv;
        int kappa = (int)floorf(u);
        float d  = u - (float)kappa;
        float om = 1.0f - d;
        const float* zs = Z + (size_t)snd[i] * (2 * N_CP);
        const float* zr = Z + (size_t)rcv[i] * (2 * N_CP);
        float sx0 = zs[kappa],        sy0 = zs[N_CP + kappa];
        float sx1 = zs[kappa + 1],    sy1 = zs[N_CP + kappa + 1];
        float rx0 = zr[kappa],        ry0 = zr[N_CP + kappa];
        float rx1 = zr[kappa + 1],    ry1 = zr[N_CP + kappa + 1];
        float cc = sx0 * rx0 + sy0 * ry0;
        float aa = sx1 * rx1 + sy1 * ry1;
        float bb = sx0 * rx1 + sy0 * ry1 + sx1 * rx0 + sy1 * ry0;
        v = beta[0] + om * om * cc + d * om * bb + d * d * aa;
    }
    __shared__ float red[256];
    red[threadIdx.x] = v;
    __syncthreads();
    for (int s = 128; s > 0; s >>= 1) {
        if (threadIdx.x < s) red[threadIdx.x] += red[threadIdx.x + s];
        __syncthreads();
    }
    if (threadIdx.x == 0) partials[blockIdx.x] = red[0];
}

// ---------------------------------------------------------------------------
// Integral: block = (tau, sgroup). tau = segment k * 65 + simpson j.
// P(t) = (1-t)*Z[:,:,k] + t*Z[:,:,k+1];  exponent = P_s . P_r  (d = 2).
// A operand pre-scaled by log2e so the elementwise op is a raw v_exp_f32.
// No per-element masking: accumulate everything, then subtract the analytic
// corrections (diagonal exp(|P_s|^2) + exp(0)=1 for padded sender rows).
// ---------------------------------------------------------------------------
__global__ void integral_kernel(const float* __restrict__ Z,
                                const int* __restrict__ nodes,
                                int fs, int bs, int sgroups,
                                float* __restrict__ partials) {
    const int tau = blockIdx.x / sgroups;
    const int sg  = blockIdx.x - tau * sgroups;
    const int k = tau / (N_SIMPSON + 1);
    const int j = tau - k * (N_SIMPSON + 1);
    const float t  = (float)j * (1.0f / (float)N_SIMPSON);
    const float om = 1.0f - t;
    const float w  = ((j == 0) || (j == N_SIMPSON)) ? (1.0f / (3.0f * N_SIMPSON))
                   : ((j & 1) ? (4.0f / (3.0f * N_SIMPSON))
                              : (2.0f / (3.0f * N_SIMPSON)));

    const int fsc     = (fs > MAX_FS) ? MAX_FS : fs;
    const int ntile16 = fsc & ~15;          // receivers covered by WMMA tiles
    const int rem     = fsc - ntile16;      // scalar tail (0 for fs=4000)

#if HAVE_EXP2
    const float ascale = LOG2E_F;           // fold exp's log2e into A operand
#else
    const float ascale = 1.0f;              // __expf fallback scales internally
#endif

    __shared__ float2 Pxy[MAX_FS];
    __shared__ float  wavePart[8];

    // Interpolate all receiver positions at time t into LDS (float2 -> b64).
    for (int i = threadIdx.x; i < fsc; i += blockDim.x) {
        const float* zp = Z + (size_t)i * (2 * N_CP);
        float2 p;
        p.x = om * zp[k]        + t * zp[k + 1];
        p.y = om * zp[N_CP + k] + t * zp[N_CP + k + 1];
        Pxy[i] = p;
    }
    __syncthreads();

    const int wave = threadIdx.x >> 5;
    const int lane = threadIdx.x & 31;
    const int s_base = (sg * 8 + wave) * 16;

    float acc = 0.0f;
    if (s_base < bs) {  // wave-uniform branch: EXEC stays full for WMMA
        // A operand, 16x4 f32 layout: lanes 0-15 carry K0=x,K1=y (scaled);
        // lanes 16-31 (K2,K3) are exact zeros -> they annihilate whatever
        // finite B values sit in lanes 16-31, so B needs no masking.
        float ax0 = 0.0f, ay0 = 0.0f;       // unscaled copies (corrections)
        int   sgid = -1;                    // global node id of this sender
        v2f a; a.x = 0.0f; a.y = 0.0f;
        if (lane < 16) {
            int s = s_base + lane;
            if (s < bs) {
                sgid = nodes[s];
                const float* zp = Z + (size_t)sgid * (2 * N_CP);
                ax0 = om * zp[k]        + t * zp[k + 1];
                ay0 = om * zp[N_CP + k] + t * zp[N_CP + k + 1];
                a.x = ascale * ax0;
                a.y = ascale * ay0;
            }
        }
#if !HAVE_WMMA_F32X4
        // Fallback: per-lane sender rows matching the D layout
        float sax[8], say[8];
        const int hi8 = (lane >> 4) * 8;
        #pragma unroll
        for (int jj = 0; jj < 8; ++jj) {
            sax[jj] = 0.0f; say[jj] = 0.0f;
            int s = s_base + jj + hi8;
            if (s < bs) {
                const float* zp = Z + (size_t)nodes[s] * (2 * N_CP);
                float px = om * zp[k]        + t * zp[k + 1];
                float py = om * zp[N_CP + k] + t * zp[N_CP + k + 1];
                sax[jj] = ascale * px;
                say[jj] = ascale * py;
            }
        }
#endif
        const int nloc = lane & 15;
        const int r_tiles = ntile16 >> 4;
        v8f accv = {};
        #pragma unroll 2
        for (int rt = 0; rt < r_tiles; ++rt) {
            const int r_id = (rt << 4) + nloc;
            float2 p = Pxy[r_id];           // all lanes: K2/K3 cancelled by A=0
            v2f b; b.x = p.x; b.y = p.y;
            v8f c = {};
#if HAVE_WMMA_F32X4
            c = __builtin_amdgcn_wmma_f32_16x16x4_f32(
                    false, a, false, b, (short)0, c, false, false);
#else
            #pragma unroll
            for (int jj = 0; jj < 8; ++jj)
                c[jj] = sax[jj] * b.x + say[jj] * b.y;
#endif
            #pragma unroll
            for (int jj = 0; jj < 8; ++jj)
                accv[jj] += fast_exp2(c[jj]);
        }
        #pragma unroll
        for (int jj = 0; jj < 8; ++jj) acc += accv[jj];

        // Analytic corrections for the unmasked tile sum:
        //  - padded sender row (s >= bs): contributed exp(0)=1 per receiver
        //  - valid sender, diagonal r == nodes[s] inside tiled region:
        //    contributed exp(|P_s|^2)
        if (lane < 16) {
            int s = s_base + lane;
            if (s >= bs) {
                acc -= (float)ntile16;
            } else if (sgid < ntile16) {
                acc -= fast_exp(ax0 * ax0 + ay0 * ay0);
            }
        }

        // Scalar tail for receivers not covered by full tiles (rem < 16).
        for (int rr = ntile16; rr < ntile16 + rem; ++rr) {
            float2 p = Pxy[rr];
            float e = fast_exp(ax0 * p.x + ay0 * p.y);
            if (lane < 16 && sgid >= 0 && sgid != rr) acc += e;
        }
    }

    #pragma unroll
    for (int off = 16; off > 0; off >>= 1)
        acc += __shfl_down(acc, off);
    if (lane == 0) wavePart[wave] = acc;
    __syncthreads();
    if (threadIdx.x == 0) {
        float s = 0.0f;
        for (int i = 0; i < 8; ++i) s += wavePart[i];
        partials[blockIdx.x] = w * s;   // fold Simpson weight here
    }
}

// ---------------------------------------------------------------------------
// Final combine: out = (fs/bs) * (10*prior - sum(fllt) + e^b*seg_len*acc/2)
// ---------------------------------------------------------------------------
__global__ void combine_kernel(const float* __restrict__ prior_slot,
                               const float* __restrict__ int_part, int n_int,
                               const float* __restrict__ fllt_part, int n_fllt,
                               const float* __restrict__ beta,
                               int fs, int bs, float seg_len,
                               float* __restrict__ out) {
    float ai = 0.0f, af = 0.0f;
    for (int i = threadIdx.x; i < n_int;  i += 256) ai += int_part[i];
    for (int i = threadIdx.x; i < n_fllt; i += 256) af += fllt_part[i];
    __shared__ float r1[256], r2[256];
    r1[threadIdx.x] = ai;
    r2[threadIdx.x] = af;
    __syncthreads();
    for (int s = 128; s > 0; s >>= 1) {
        if (threadIdx.x < s) {
            r1[threadIdx.x] += r1[threadIdx.x + s];
            r2[threadIdx.x] += r2[threadIdx.x + s];
        }
        __syncthreads();
    }
    if (threadIdx.x == 0) {
        float integral = fast_exp(beta[0]) * seg_len * r1[0] * 0.5f;
        float val = PENALTY_F * prior_slot[0] - r2[0] + integral;
        out[0] = ((float)fs / (float)bs) * val;
    }
}

// ---------------------------------------------------------------------------
extern "C" void kernel_launch(void* const* d_in, const int* in_sizes, int n_in,
                              void* d_out, int out_size, void* d_ws, size_t ws_size,
                              hipStream_t stream) {
    const float* Z     = (const float*)d_in[0];
    const float* beta  = (const float*)d_in[1];
    const float* ts    = (const float*)d_in[2];
    const int*   snd   = (const int*)d_in[3];
    const int*   rcv   = (const int*)d_in[4];
    const int*   nodes = (const int*)d_in[5];

    const int fs       = in_sizes[0] / (2 * N_CP);   // 4000
    const int n_events = in_sizes[2];                // 200000
    const int bs       = in_sizes[5];                // 1000
    const float seg_len = (1.0f + 0.0001f) / (float)(N_CP - 1);

    const int n_tau   = N_SEG * (N_SIMPSON + 1);     // 1040 time samples
    const int s_tiles = (bs + 15) / 16;              // 63
    const int sgroups = (s_tiles + 7) / 8;           // 8 blocks per tau
    const int n_int_blocks  = n_tau * sgroups;       // 8320
    const int n_fllt_blocks = (n_events + 255) / 256;

    float* ws         = (float*)d_ws;
    float* prior_slot = ws;                      // [1]
    float* int_part   = ws + 1;                  // [n_int_blocks]
    float* fllt_part  = int_part + n_int_blocks; // [n_fllt_blocks]

    prior_kernel<<<1, 256, 0, stream>>>(Z, nodes, bs, prior_slot);
    fllt_kernel<<<n_fllt_blocks, 256, 0, stream>>>(Z, beta, ts, snd, rcv,
                                                   n_events, 1.0f / seg_len,
                                                   fllt_part);
    integral_kernel<<<n_int_blocks, 256, 0, stream>>>(Z, nodes, fs, bs, sgroups,
                                                      int_part);
    combine_kernel<<<1, 256, 0, stream>>>(prior_slot, int_part, n_int_blocks,
                                          fllt_part, n_fllt_blocks, beta,
                                          fs, bs, seg_len, (float*)d_out);
}